// HGDC_C_20925080666231
// MI455X (gfx1250) — compile-verified
//
#include <hip/hip_runtime.h>
#include <hip/hip_bf16.h>

#define N_NODES 50000
#define N_EDGES 800000
#define HDIM    64
// LDS layout: W element (k,n) -> sW[(k>>1)*PSTR + 2n + (k&1)].
// PSTR=160 floats: half-wave offset 160 % 64 == 32 -> disjoint 32-bank windows.
#define PSTR    160

typedef __attribute__((ext_vector_type(2))) float v2f;
typedef __attribute__((ext_vector_type(8))) float v8f;

// ---------------------------------------------------------------------------
// utility kernels
// ---------------------------------------------------------------------------
__global__ void zero_f32(float* __restrict__ p, int n) {
    int i = blockIdx.x * blockDim.x + threadIdx.x;
    if (i < n) p[i] = 0.0f;
}

__global__ void degree_kernel(const int* __restrict__ col, float* __restrict__ deg, int e) {
    int i = blockIdx.x * blockDim.x + threadIdx.x;
    if (i < e) unsafeAtomicAdd(&deg[col[i]], 1.0f);
}

__global__ void dinv_kernel(float* __restrict__ deg, int n) {
    int i = blockIdx.x * blockDim.x + threadIdx.x;
    if (i < n) {
        float d = deg[i];
        deg[i] = (d > 0.0f) ? rsqrtf(fmaxf(d, 1e-12f)) : 0.0f;
    }
}

__global__ void norm_kernel(const int* __restrict__ row, const int* __restrict__ col,
                            const float* __restrict__ dinv, float* __restrict__ norm, int e) {
    int i = blockIdx.x * blockDim.x + threadIdx.x;
    if (i < e) norm[i] = dinv[row[i]] * dinv[col[i]];
}

// out[n, colOff + h] = bias[h]   (prepare accumulation target for scatter-add)
__global__ void init_bias_kernel(float* __restrict__ out, const float* __restrict__ bias,
                                 int stride, int colOff, int n) {
    int i = blockIdx.x * blockDim.x + threadIdx.x;
    if (i < n * HDIM) {
        int node = i >> 6;
        int h    = i & 63;
        out[(size_t)node * stride + colOff + h] = bias[h];
    }
}

// ---------------------------------------------------------------------------
// WMMA fp32 GEMM:  Y[nrows,64] = X[nrows,K] @ W[K,64]  (+bias) (+relu)
// W staged per-workgroup into LDS, K-pair interleaved so each lane's B
// fragment is a single aligned ds_load_b64 (no register shuffles, no
// bank conflicts). One wave computes a 16x64 tile: 4 accumulators of
// 16x16, K stepped by 4 with V_WMMA_F32_16X16X4_F32.
//   A 16x4 layout: lanes 0-15 -> M=lane, VGPR{0,1}=K{0,1}; lanes 16-31 -> K{2,3}
//   B 4x16 layout: lanes 0-15 -> N=lane, VGPR{0,1}=K{0,1}; lanes 16-31 -> K{2,3}
//   C/D 16x16:     VGPR v, lanes 0-15 -> M=v, N=lane; lanes 16-31 -> M=v+8
// ---------------------------------------------------------------------------
#define GF_BIAS 1
#define GF_RELU 2

__global__ __launch_bounds__(256)
void gemm_wmma_f32(const float* __restrict__ X, const float* __restrict__ W,
                   const float* __restrict__ bias, float* __restrict__ Y,
                   int nrows, int K, int flags) {
    __shared__ float sW[64 * PSTR];                  // up to K=128 (64 k-pairs), 40 KB

    // ---- cooperative stage of W into LDS, pair-interleaved ----------------
    {
        int pairs = (K >> 1) * HDIM;                 // 2048 or 4096
        for (int idx = threadIdx.x; idx < pairs; idx += blockDim.x) {
            int kp = idx >> 6;                       // k-pair row
            int n  = idx & 63;
            v2f p;
            p.x = W[(size_t)(2 * kp)     * HDIM + n];   // coalesced across n
            p.y = W[(size_t)(2 * kp + 1) * HDIM + n];
            *(v2f*)(sW + kp * PSTR + (n << 1)) = p;     // 8B-aligned
        }
    }
    __syncthreads();                                 // ALL waves reach this barrier

    int wave = (blockIdx.x * blockDim.x + threadIdx.x) >> 5;
    int lane = threadIdx.x & 31;
    int ntiles = nrows >> 4;                         // nrows is a multiple of 16
    if (wave >= ntiles) return;                      // wave-uniform: EXEC stays all-ones

    int m0   = wave << 4;
    int half = lane >> 4;                            // 0: lanes 0-15, 1: lanes 16-31
    int l16  = lane & 15;

    v8f acc0 = {}, acc1 = {}, acc2 = {}, acc3 = {};

    const float* xrow  = X + (size_t)(m0 + l16) * K + (half << 1);
    const float* wbase = sW + half * PSTR + (l16 << 1);   // k-pair row (k>>1)+half
    for (int k = 0; k < K; k += 4) {
        v2f a = *(const v2f*)(xrow + k);

        const float* wr = wbase + (k >> 1) * PSTR;
        v2f b0 = *(const v2f*)(wr + 0);               // cols  0..15
        v2f b1 = *(const v2f*)(wr + 32);              // cols 16..31
        v2f b2 = *(const v2f*)(wr + 64);              // cols 32..47
        v2f b3 = *(const v2f*)(wr + 96);              // cols 48..63

        acc0 = __builtin_amdgcn_wmma_f32_16x16x4_f32(false, a, false, b0, (short)0, acc0, false, false);
        acc1 = __builtin_amdgcn_wmma_f32_16x16x4_f32(false, a, false, b1, (short)0, acc1, false, false);
        acc2 = __builtin_amdgcn_wmma_f32_16x16x4_f32(false, a, false, b2, (short)0, acc2, false, false);
        acc3 = __builtin_amdgcn_wmma_f32_16x16x4_f32(false, a, false, b3, (short)0, acc3, false, false);
    }

    #pragma unroll
    for (int v = 0; v < 8; ++v) {
        int r = m0 + v + (half << 3);
        float* yr = Y + (size_t)r * HDIM + l16;
        float o0 = acc0[v], o1 = acc1[v], o2 = acc2[v], o3 = acc3[v];
        if (flags & GF_BIAS) {
            o0 += bias[l16];
            o1 += bias[16 + l16];
            o2 += bias[32 + l16];
            o3 += bias[48 + l16];
        }
        if (flags & GF_RELU) {
            o0 = fmaxf(o0, 0.0f); o1 = fmaxf(o1, 0.0f);
            o2 = fmaxf(o2, 0.0f); o3 = fmaxf(o3, 0.0f);
        }
        yr[0]  = o0;
        yr[16] = o1;
        yr[32] = o2;
        yr[48] = o3;
    }
}

// ---------------------------------------------------------------------------
// scatter: Out[col[e], colOff + h] += Y[row[e], h] * norm[e]
// one wave per edge; 2 features per lane (64 feats / 32 lanes, wave32)
// ---------------------------------------------------------------------------
__global__ void scatter_kernel(const float* __restrict__ Y,
                               const int* __restrict__ row, const int* __restrict__ col,
                               const float* __restrict__ norm,
                               float* __restrict__ Out, int stride, int colOff, int e) {
    int edge = (blockIdx.x * blockDim.x + threadIdx.x) >> 5;
    if (edge >= e) return;
    int lane = threadIdx.x & 31;
    int r = row[edge], c = col[edge];
    float nv = norm[edge];
    float2 yv = *(const float2*)(Y + (size_t)r * HDIM + (lane << 1));
    float* op = Out + (size_t)c * stride + colOff + (lane << 1);
    unsafeAtomicAdd(op,     yv.x * nv);
    unsafeAtomicAdd(op + 1, yv.y * nv);
}

// ---------------------------------------------------------------------------
// readout: out[n] = (R0@Wr0+br0)*w0 + (R1@Wr1+br1)*w1 + (R2@Wr2+br2)*w2 + (R3@Wr3+br3)*w3
// one wave per node, shuffle reduction
// ---------------------------------------------------------------------------
__global__ void readout_kernel(const float* __restrict__ R0, const float* __restrict__ R1,
                               const float* __restrict__ R2, const float* __restrict__ R3,
                               const float* __restrict__ Wr0, const float* __restrict__ br0,
                               const float* __restrict__ Wr1, const float* __restrict__ br1,
                               const float* __restrict__ Wr2, const float* __restrict__ br2,
                               const float* __restrict__ Wr3, const float* __restrict__ br3,
                               const float* __restrict__ w0, const float* __restrict__ w1,
                               const float* __restrict__ w2, const float* __restrict__ w3,
                               float* __restrict__ out, int n) {
    int node = (blockIdx.x * blockDim.x + threadIdx.x) >> 5;
    if (node >= n) return;
    int lane = threadIdx.x & 31;

    float c0 = w0[0], c1 = w1[0], c2 = w2[0], c3 = w3[0];

    const float* r0 = R0 + (size_t)node * 64;
    const float* r1 = R1 + (size_t)node * 128;
    const float* r2 = R2 + (size_t)node * 128;
    const float* r3 = R3 + (size_t)node * 128;

    float s0 = r0[lane] * Wr0[lane] + r0[lane + 32] * Wr0[lane + 32];
    float s1 = 0.0f, s2 = 0.0f, s3 = 0.0f;
    #pragma unroll
    for (int i = 0; i < 4; ++i) {
        int idx = lane + i * 32;
        s1 += r1[idx] * Wr1[idx];
        s2 += r2[idx] * Wr2[idx];
        s3 += r3[idx] * Wr3[idx];
    }
    float t = s0 * c0 + s1 * c1 + s2 * c2 + s3 * c3;
    #pragma unroll
    for (int off = 16; off >= 1; off >>= 1)
        t += __shfl_xor(t, off, 32);
    if (lane == 0)
        out[node] = t + br0[0] * c0 + br1[0] * c1 + br2[0] * c2 + br3[0] * c3;
}

// ---------------------------------------------------------------------------
// host orchestration
// ---------------------------------------------------------------------------
extern "C" void kernel_launch(void* const* d_in, const int* in_sizes, int n_in,
                              void* d_out, int out_size, void* d_ws, size_t ws_size,
                              hipStream_t stream) {
    const float* x    = (const float*)d_in[0];
    const int*   ei1  = (const int*)d_in[1];
    const int*   ei2  = (const int*)d_in[2];
    const float* W_in = (const float*)d_in[3];
    const float* b_in = (const float*)d_in[4];
    const float* W11  = (const float*)d_in[5];
    const float* b11  = (const float*)d_in[6];
    const float* W12  = (const float*)d_in[7];
    const float* b12  = (const float*)d_in[8];
    const float* W21  = (const float*)d_in[9];
    const float* b21  = (const float*)d_in[10];
    const float* W22  = (const float*)d_in[11];
    const float* b22  = (const float*)d_in[12];
    const float* W31  = (const float*)d_in[13];
    const float* b31  = (const float*)d_in[14];
    const float* W32  = (const float*)d_in[15];
    const float* b32  = (const float*)d_in[16];
    const float* Wr0  = (const float*)d_in[17];
    const float* br0  = (const float*)d_in[18];
    const float* Wr1  = (const float*)d_in[19];
    const float* br1  = (const float*)d_in[20];
    const float* Wr2  = (const float*)d_in[21];
    const float* br2  = (const float*)d_in[22];
    const float* Wr3  = (const float*)d_in[23];
    const float* br3  = (const float*)d_in[24];
    const float* w0   = (const float*)d_in[25];
    const float* w1   = (const float*)d_in[26];
    const float* w2   = (const float*)d_in[27];
    const float* w3   = (const float*)d_in[28];

    const int* r1 = ei1;             const int* c1 = ei1 + N_EDGES;
    const int* r2 = ei2;             const int* c2 = ei2 + N_EDGES;

    // workspace carve-out (256B aligned regions)
    auto align256 = [](size_t b) { return (b + 255) & ~(size_t)255; };
    char* w = (char*)d_ws;
    float* deg1  = (float*)w; w += align256((size_t)N_NODES * 4);
    float* deg2  = (float*)w; w += align256((size_t)N_NODES * 4);
    float* norm1 = (float*)w; w += align256((size_t)N_EDGES * 4);
    float* norm2 = (float*)w; w += align256((size_t)N_EDGES * 4);
    float* R0    = (float*)w; w += align256((size_t)N_NODES * 64 * 4);
    float* Yt    = (float*)w; w += align256((size_t)N_NODES * 64 * 4);
    float* R1    = (float*)w; w += align256((size_t)N_NODES * 128 * 4);
    float* R2    = (float*)w; w += align256((size_t)N_NODES * 128 * 4);
    float* R3    = (float*)w; w += align256((size_t)N_NODES * 128 * 4);
    (void)ws_size; (void)n_in; (void)in_sizes; (void)out_size;

    const int TB = 256;
    dim3 blk(TB);
    int gNodes   = (N_NODES + TB - 1) / TB;
    int gEdges   = (N_EDGES + TB - 1) / TB;
    int gEdgeWav = ((N_EDGES * 32) + TB - 1) / TB;          // one wave per edge
    int gNodeWav = ((N_NODES * 32) + TB - 1) / TB;          // one wave per node
    int gInit    = ((N_NODES * HDIM) + TB - 1) / TB;
    int gGemm    = ((N_NODES / 16) + (TB / 32) - 1) / (TB / 32); // 16-row tiles, 8 waves/block

    // ---- edge norms --------------------------------------------------------
    zero_f32<<<gNodes, blk, 0, stream>>>(deg1, N_NODES);
    zero_f32<<<gNodes, blk, 0, stream>>>(deg2, N_NODES);
    degree_kernel<<<gEdges, blk, 0, stream>>>(c1, deg1, N_EDGES);
    degree_kernel<<<gEdges, blk, 0, stream>>>(c2, deg2, N_EDGES);
    dinv_kernel<<<gNodes, blk, 0, stream>>>(deg1, N_NODES);
    dinv_kernel<<<gNodes, blk, 0, stream>>>(deg2, N_NODES);
    norm_kernel<<<gEdges, blk, 0, stream>>>(r1, c1, deg1, norm1, N_EDGES);
    norm_kernel<<<gEdges, blk, 0, stream>>>(r2, c2, deg2, norm2, N_EDGES);

    // ---- R0 = relu(x @ W_in + b_in) ---------------------------------------
    gemm_wmma_f32<<<gGemm, blk, 0, stream>>>(x, W_in, b_in, R0, N_NODES, 64, GF_BIAS | GF_RELU);

    // ---- layer 1: R1 = [gcn(R0,W11,e1) | gcn(R0,W12,e2)] -------------------
    gemm_wmma_f32<<<gGemm, blk, 0, stream>>>(R0, W11, nullptr, Yt, N_NODES, 64, 0);
    init_bias_kernel<<<gInit, blk, 0, stream>>>(R1, b11, 128, 0, N_NODES);
    scatter_kernel<<<gEdgeWav, blk, 0, stream>>>(Yt, r1, c1, norm1, R1, 128, 0, N_EDGES);

    gemm_wmma_f32<<<gGemm, blk, 0, stream>>>(R0, W12, nullptr, Yt, N_NODES, 64, 0);
    init_bias_kernel<<<gInit, blk, 0, stream>>>(R1, b12, 128, 64, N_NODES);
    scatter_kernel<<<gEdgeWav, blk, 0, stream>>>(Yt, r2, c2, norm2, R1, 128, 64, N_EDGES);

    // ---- layer 2: R2 = [gcn(R1,W21,e1) | gcn(R1,W22,e2)] -------------------
    gemm_wmma_f32<<<gGemm, blk, 0, stream>>>(R1, W21, nullptr, Yt, N_NODES, 128, 0);
    init_bias_kernel<<<gInit, blk, 0, stream>>>(R2, b21, 128, 0, N_NODES);
    scatter_kernel<<<gEdgeWav, blk, 0, stream>>>(Yt, r1, c1, norm1, R2, 128, 0, N_EDGES);

    gemm_wmma_f32<<<gGemm, blk, 0, stream>>>(R1, W22, nullptr, Yt, N_NODES, 128, 0);
    init_bias_kernel<<<gInit, blk, 0, stream>>>(R2, b22, 128, 64, N_NODES);
    scatter_kernel<<<gEdgeWav, blk, 0, stream>>>(Yt, r2, c2, norm2, R2, 128, 64, N_EDGES);

    // ---- layer 3: R3 = [gcn(R2,W31,e1) | gcn(R2,W32,e2)] -------------------
    gemm_wmma_f32<<<gGemm, blk, 0, stream>>>(R2, W31, nullptr, Yt, N_NODES, 128, 0);
    init_bias_kernel<<<gInit, blk, 0, stream>>>(R3, b31, 128, 0, N_NODES);
    scatter_kernel<<<gEdgeWav, blk, 0, stream>>>(Yt, r1, c1, norm1, R3, 128, 0, N_EDGES);

    gemm_wmma_f32<<<gGemm, blk, 0, stream>>>(R2, W32, nullptr, Yt, N_NODES, 128, 0);
    init_bias_kernel<<<gInit, blk, 0, stream>>>(R3, b32, 128, 64, N_NODES);
    scatter_kernel<<<gEdgeWav, blk, 0, stream>>>(Yt, r2, c2, norm2, R3, 128, 64, N_EDGES);

    // ---- readout -----------------------------------------------------------
    readout_kernel<<<gNodeWav, blk, 0, stream>>>(R0, R1, R2, R3,
                                                 Wr0, br0, Wr1, br1, Wr2, br2, Wr3, br3,
                                                 w0, w1, w2, w3,
                                                 (float*)d_out, N_NODES);
}